// ObstacleHead_77120432767342
// MI455X (gfx1250) — compile-verified
//
#include <hip/hip_runtime.h>
#include <hip/hip_bf16.h>

// Problem constants (match reference)
#define B_     16
#define N_     20
#define H_     224
#define W_     224
#define HW_    (H_*W_)
#define HID_   1024
#define HEADS_ 4
#define HD_    256          // HID/HEADS
#define BN_    (B_*N_)      // 320
#define BBH_   2048         // backbone hidden

typedef float v2f __attribute__((ext_vector_type(2)));
typedef float v8f __attribute__((ext_vector_type(8)));

// ---------------------------------------------------------------------------
// Kernel 1: fused mask statistics.  One block per (b,n).
// Produces: overlap(b,n), iou(b,n) -> edge(B, 2N);  f(b,n)=mean(obj);  objsum(b,n)
// ---------------------------------------------------------------------------
__global__ void obh_mask_stats(const float* __restrict__ obj,   // (B,N,H,W)
                               const float* __restrict__ tgt,   // (B,H,W)
                               const float* __restrict__ bboxes,// (B,N,4)
                               float* __restrict__ edge,        // (B, 2N)
                               float* __restrict__ f,           // (B*N)
                               float* __restrict__ objsum)      // (B*N)
{
    int bn  = blockIdx.x;
    int b   = bn / N_;
    int tid = threadIdx.x;
    const float* om = obj + (size_t)bn * HW_;
    const float* tm = tgt + (size_t)b  * HW_;

    int ix1 = (int)floorf(bboxes[bn*4+0]);
    int iy1 = (int)floorf(bboxes[bn*4+1]);
    int ix2 = (int)floorf(bboxes[bn*4+2]);
    int iy2 = (int)floorf(bboxes[bn*4+3]);

    float so = 0.f, sov = 0.f, st = 0.f, si = 0.f;
    for (int i = tid; i < HW_; i += blockDim.x) {
        int y = i / W_;
        int x = i - y * W_;
        float o = om[i];
        float t = tm[i];
        so  += o;
        sov += o * t;
        st  += t;
        if (y >= iy1 && y < iy2 && x >= ix1 && x < ix2) si += t;
    }

    __shared__ float4 red[256];
    red[tid] = make_float4(so, sov, st, si);
    __syncthreads();
    for (int s = 128; s > 0; s >>= 1) {
        if (tid < s) {
            red[tid].x += red[tid+s].x;
            red[tid].y += red[tid+s].y;
            red[tid].z += red[tid+s].z;
            red[tid].w += red[tid+s].w;
        }
        __syncthreads();
    }

    if (tid == 0) {
        float osum    = red[0].x;
        float overlap = red[0].y;
        float tarea   = red[0].z;
        float inter   = red[0].w;
        int ry1 = iy1 > 0 ? iy1 : 0;
        int ry2 = iy2 < H_ ? iy2 : H_;
        int rx1 = ix1 > 0 ? ix1 : 0;
        int rx2 = ix2 < W_ ? ix2 : W_;
        float rowc = (float)(ry2 > ry1 ? ry2 - ry1 : 0);
        float colc = (float)(rx2 > rx1 ? rx2 - rx1 : 0);
        float box_area = rowc * colc;
        float iou = inter / (box_area + tarea - inter + 1e-8f);
        int n = bn - b * N_;
        edge[b*(2*N_) + 2*n + 0] = overlap;
        edge[b*(2*N_) + 2*n + 1] = iou;
        objsum[bn] = osum;
        f[bn] = osum * (1.0f / (float)HW_);
    }
}

// ---------------------------------------------------------------------------
// Kernel 2: backbone hidden layer.  f3 @ w1 == f * (w1[0]+w1[1]+w1[2]) per col.
// h(r,c) = relu(f[r]*colsum_w1[c] + b1[c]),  (320 x 2048)
// ---------------------------------------------------------------------------
__global__ void obh_bb_hidden(const float* __restrict__ f,
                              const float* __restrict__ w1,   // (3,2048)
                              const float* __restrict__ b1,   // (2048)
                              float* __restrict__ h)          // (320,2048)
{
    int i = blockIdx.x * blockDim.x + threadIdx.x;
    if (i >= BN_ * BBH_) return;
    int r = i / BBH_;
    int c = i - r * BBH_;
    float s = w1[c] + w1[BBH_ + c] + w1[2*BBH_ + c];
    h[i] = fmaxf(f[r] * s + b1[c], 0.f);
}

// ---------------------------------------------------------------------------
// Kernel 3: generic f32 WMMA GEMM.  C(M,Nn) = A(M,K) @ B(K,Nn) [+ epilogue]
// One wave computes one 16x16 output tile via v_wmma_f32_16x16x4_f32.
// K unrolled x16 per iteration: 4x global_load_b64 (A pairs, via v2f base) +
// 8x global_load_b32 (B), all in SADDR + 32-bit voffset scale_offset form,
// clause-issued; 4 WMMAs on two alternating accumulators.  Unconditional
// speculative prefetch of the next B rows (ISA 10.5: invalid addresses are
// silently dropped, no guard branch needed).
// NT=true issues B loads with the non-temporal cache hint (for the two 84MB
// single-use weight streams).
// Lane layouts per CDNA5 ISA 7.12.2:
//   A 16x4 : lanes 0-15 M=lane {K=k,k+1}; lanes 16-31 M=lane-16 {K=k+2,k+3}
//   B 4x16 : lanes 0-15 N=lane {K=k,k+1}; lanes 16-31 N=lane-16 {K=k+2,k+3}
//   C 16x16: vgpr r -> M = m0 + r + 8*(lane>=16), N = n0 + (lane&15)
// EPI: 0 = +bias, 1 = relu(bn(+bias)), 2 = atomic split-K accumulate (raw)
// Requirements honored by all launches: lda, k0 even; M,Nn multiples of 16.
// ---------------------------------------------------------------------------
#define WMMA_F32(AA, BB, CC) \
    __builtin_amdgcn_wmma_f32_16x16x4_f32(false, (AA), false, (BB), (short)0, (CC), false, false)

template<int EPI, bool NT>
__global__ void obh_gemm_wmma(const float* __restrict__ A, int lda,
                              const float* __restrict__ Bw, int ldb,
                              const float* __restrict__ bias,
                              const float* __restrict__ g,  const float* __restrict__ be,
                              const float* __restrict__ mu, const float* __restrict__ var,
                              float* __restrict__ C, int ldc,
                              int M, int Nn, int K, int kChunk)
{
    int wavesPerBlock = blockDim.x >> 5;
    int wave = threadIdx.x >> 5;
    int lane = threadIdx.x & 31;
    int tileN = blockIdx.x * wavesPerBlock + wave;
    int tileM = blockIdx.y;
    int n0 = tileN << 4;
    int m0 = tileM << 4;
    if (n0 >= Nn || m0 >= M) return;   // wave-uniform; grids tile exactly anyway

    int half = lane >> 4;      // K sub-pair select (0 -> k,k+1 ; 1 -> k+2,k+3)
    int lidx = lane & 15;

    int k0   = blockIdx.z * kChunk;
    int kend = k0 + kChunk;
    if (kend > K) kend = K;

    v8f acc0 = {0.f,0.f,0.f,0.f,0.f,0.f,0.f,0.f};
    v8f acc1 = {0.f,0.f,0.f,0.f,0.f,0.f,0.f,0.f};

    int k = k0;

    // 32-bit running element offsets against the uniform A / Bw bases.
    // Max values: A < 0.7M elements, B < 21M elements  -->  fits u32 easily.
    unsigned uldb  = (unsigned)ldb;
    unsigned bstep = 16u * uldb;
    unsigned kk0   = (unsigned)(k0 + (half << 1));
    unsigned col   = (unsigned)(n0 + lidx);
    unsigned o0  = kk0 * uldb + col;
    unsigned o1  = o0 + uldb;
    unsigned o4  = o0 + 4u*uldb;
    unsigned o5  = o0 + 5u*uldb;
    unsigned o8  = o0 + 8u*uldb;
    unsigned o9  = o0 + 9u*uldb;
    unsigned o12 = o0 + 12u*uldb;
    unsigned o13 = o0 + 13u*uldb;

    // A indexed as float2 pairs: aoff2 in v2f units (lda, k0 always even).
    const v2f* A2 = (const v2f*)A;
    unsigned aoff2 = (unsigned)(m0 + lidx) * (unsigned)(lda >> 1)
                   + (unsigned)(k0 >> 1) + (unsigned)half;

    // Main loop: 16 K-values per iteration (4 WMMAs, 12 loads in flight).
    for (; k + 16 <= kend; k += 16) {
        v2f a0, a1, a2, a3, b0, b1, b2, b3;
        a0 = A2[aoff2 + 0];
        a1 = A2[aoff2 + 2];
        a2 = A2[aoff2 + 4];
        a3 = A2[aoff2 + 6];
        if (NT) {
            b0.x = __builtin_nontemporal_load(&Bw[o0]);
            b0.y = __builtin_nontemporal_load(&Bw[o1]);
            b1.x = __builtin_nontemporal_load(&Bw[o4]);
            b1.y = __builtin_nontemporal_load(&Bw[o5]);
            b2.x = __builtin_nontemporal_load(&Bw[o8]);
            b2.y = __builtin_nontemporal_load(&Bw[o9]);
            b3.x = __builtin_nontemporal_load(&Bw[o12]);
            b3.y = __builtin_nontemporal_load(&Bw[o13]);
        } else {
            b0.x = Bw[o0];  b0.y = Bw[o1];
            b1.x = Bw[o4];  b1.y = Bw[o5];
            b2.x = Bw[o8];  b2.y = Bw[o9];
            b3.x = Bw[o12]; b3.y = Bw[o13];
        }
        // Unconditional speculative prefetch of next iteration's B rows:
        // invalid/past-end addresses are dropped silently (ISA 10.5).
        __builtin_prefetch(&Bw[o0 + bstep], 0, 0);
        __builtin_prefetch(&Bw[o8 + bstep], 0, 0);
        aoff2 += 8u;
        o0 += bstep; o1 += bstep; o4  += bstep; o5  += bstep;
        o8 += bstep; o9 += bstep; o12 += bstep; o13 += bstep;
        acc0 = WMMA_F32(a0, b0, acc0);
        acc1 = WMMA_F32(a1, b1, acc1);
        acc0 = WMMA_F32(a2, b2, acc0);
        acc1 = WMMA_F32(a3, b3, acc1);
    }
    // Remainder: 4 K-values per step (only the K=40 edge GEMM hits this).
    for (; k < kend; k += 4) {
        unsigned kk = (unsigned)(k + (half << 1));
        v2f a, bv;
        a    = A2[(unsigned)(m0 + lidx) * (unsigned)(lda >> 1) + (kk >> 1)];
        bv.x = Bw[kk * uldb + col];
        bv.y = Bw[(kk + 1u) * uldb + col];
        acc0 = WMMA_F32(a, bv, acc0);
    }

    int cm = m0 + (half << 3);   // rows cm .. cm+7
    int cn = n0 + lidx;
    if (EPI == 2) {
        #pragma unroll
        for (int r = 0; r < 8; ++r)
            atomicAdd(&C[(size_t)(cm + r) * ldc + cn], acc0[r] + acc1[r]);
    } else {
        float bs = bias[cn];
        float sc = 0.f, sh = 0.f;
        if (EPI == 1) {
            sc = rsqrtf(var[cn] + 1e-5f) * g[cn];
            sh = be[cn] - mu[cn] * sc;
        }
        #pragma unroll
        for (int r = 0; r < 8; ++r) {
            float x = acc0[r] + acc1[r] + bs;
            if (EPI == 1) x = fmaxf(x * sc + sh, 0.f);
            C[(size_t)(cm + r) * ldc + cn] = x;
        }
    }
}

// ---------------------------------------------------------------------------
// Kernel 4: multi-head attention, one block per (b, head).  Tiny (20x20).
// ---------------------------------------------------------------------------
__global__ void obh_attention(const float* __restrict__ qb,
                              const float* __restrict__ kb,
                              const float* __restrict__ vb,
                              float* __restrict__ ctx)
{
    int b = blockIdx.x / HEADS_;
    int h = blockIdx.x - b * HEADS_;
    __shared__ float sq[N_ * HD_];
    __shared__ float sk[N_ * HD_];
    __shared__ float sv[N_ * HD_];
    __shared__ float sc[N_ * N_];
    int tid = threadIdx.x;

    for (int i = tid; i < N_ * HD_; i += blockDim.x) {
        int n = i / HD_, d = i - n * HD_;
        size_t off = (size_t)(b * N_ + n) * HID_ + h * HD_ + d;
        sq[i] = qb[off];
        sk[i] = kb[off];
        sv[i] = vb[off];
    }
    __syncthreads();

    for (int p = tid; p < N_ * N_; p += blockDim.x) {
        int qi = p / N_, ki = p - qi * N_;
        float acc = 0.f;
        for (int d = 0; d < HD_; ++d) acc += sq[qi*HD_ + d] * sk[ki*HD_ + d];
        sc[p] = acc * 0.0625f;          // 1/sqrt(HD)=1/16
    }
    __syncthreads();

    if (tid < N_) {
        float mx = -3.4e38f;
        for (int j = 0; j < N_; ++j) mx = fmaxf(mx, sc[tid*N_ + j]);
        float sum = 0.f;
        for (int j = 0; j < N_; ++j) { float e = __expf(sc[tid*N_ + j] - mx); sc[tid*N_ + j] = e; sum += e; }
        float inv = 1.f / sum;
        for (int j = 0; j < N_; ++j) sc[tid*N_ + j] *= inv;
    }
    __syncthreads();

    for (int i = tid; i < N_ * HD_; i += blockDim.x) {
        int n = i / HD_, d = i - n * HD_;
        float acc = 0.f;
        for (int j = 0; j < N_; ++j) acc += sc[n*N_ + j] * sv[j*HD_ + d];
        ctx[(size_t)(b * N_ + n) * HID_ + h * HD_ + d] = acc;
    }
}

// ---------------------------------------------------------------------------
// Kernel 5: fused bias + batchnorm + relu epilogue for the split-K GEMM.
// ---------------------------------------------------------------------------
__global__ void obh_bias_bn_relu(float* __restrict__ X,
                                 const float* __restrict__ bias,
                                 const float* __restrict__ g,  const float* __restrict__ be,
                                 const float* __restrict__ mu, const float* __restrict__ var,
                                 int total)
{
    int i = blockIdx.x * blockDim.x + threadIdx.x;
    if (i >= total) return;
    int c = i & (HID_ - 1);
    float x  = X[i] + bias[c];
    float sc = rsqrtf(var[c] + 1e-5f) * g[c];
    X[i] = fmaxf((x - mu[c]) * sc + be[c], 0.f);
}

// ---------------------------------------------------------------------------
// Kernel 6: final 16x20 projection + padding select.  One wave per output.
// ---------------------------------------------------------------------------
__global__ void obh_final_scores(const float* __restrict__ s,     // (16,1024)
                                 const float* __restrict__ w2,    // (1024,20)
                                 const float* __restrict__ b2,    // (20)
                                 const float* __restrict__ objsum,// (320)
                                 float* __restrict__ out)         // (16,20)
{
    int wave = threadIdx.x >> 5;
    int lane = threadIdx.x & 31;
    int o = blockIdx.x * (blockDim.x >> 5) + wave;
    if (o >= BN_) return;
    int b = o / N_;
    int j = o - b * N_;
    const float* srow = s + (size_t)b * HID_;
    float acc = 0.f;
    for (int k = lane; k < HID_; k += 32)
        acc += srow[k] * w2[(size_t)k * N_ + j];
    #pragma unroll
    for (int sh = 16; sh > 0; sh >>= 1)
        acc += __shfl_xor(acc, sh, 32);
    if (lane == 0) {
        float val = acc + b2[j];
        out[o] = (objsum[o] == 0.f) ? -1e-6f : val;
    }
}

// ---------------------------------------------------------------------------
// Host orchestration
// ---------------------------------------------------------------------------
extern "C" void kernel_launch(void* const* d_in, const int* in_sizes, int n_in,
                              void* d_out, int out_size, void* d_ws, size_t ws_size,
                              hipStream_t stream)
{
    // setup_inputs() order (scene_mask at [0] is unused by the reference)
    const float* target_mask  = (const float*)d_in[1];
    const float* object_masks = (const float*)d_in[2];
    const float* bboxes       = (const float*)d_in[3];
    const float* bb_w1 = (const float*)d_in[4];
    const float* bb_b1 = (const float*)d_in[5];
    const float* bb_fw = (const float*)d_in[6];
    const float* bb_fb = (const float*)d_in[7];
    const float* or_w1 = (const float*)d_in[8];
    const float* or_b1 = (const float*)d_in[9];
    const float* or_g  = (const float*)d_in[10];
    const float* or_be = (const float*)d_in[11];
    const float* or_m  = (const float*)d_in[12];
    const float* or_v  = (const float*)d_in[13];
    const float* or_w2 = (const float*)d_in[14];
    const float* or_b2 = (const float*)d_in[15];
    const float* at_w1 = (const float*)d_in[16];
    const float* at_b1 = (const float*)d_in[17];
    const float* at_g  = (const float*)d_in[18];
    const float* at_be = (const float*)d_in[19];
    const float* at_m  = (const float*)d_in[20];
    const float* at_v  = (const float*)d_in[21];
    const float* at_w2 = (const float*)d_in[22];
    const float* at_b2 = (const float*)d_in[23];
    const float* q_w   = (const float*)d_in[24];
    const float* q_b   = (const float*)d_in[25];
    const float* k_w   = (const float*)d_in[26];
    const float* k_b   = (const float*)d_in[27];
    const float* v_w   = (const float*)d_in[28];
    const float* v_b   = (const float*)d_in[29];
    const float* o_w   = (const float*)d_in[30];
    const float* o_b   = (const float*)d_in[31];

    float* ws = (float*)d_ws;
    size_t off = 0;
    auto walloc = [&](size_t n) { float* p = ws + off; off += n; return p; };
    float* objsum = walloc(BN_);
    float* f      = walloc(BN_);
    float* edge   = walloc((size_t)B_ * 2 * N_);
    float* hbb    = walloc((size_t)BN_ * BBH_);   // (320,2048)
    float* feats  = walloc((size_t)BN_ * HID_);   // object_feats (320,1024)
    float* rel1   = walloc((size_t)B_  * HID_);   // (16,1024)
    float* rel    = walloc((size_t)BN_ * HID_);   // (16,20480)==(320,1024)
    float* qb     = walloc((size_t)BN_ * HID_);
    float* kb     = walloc((size_t)BN_ * HID_);
    float* vb     = walloc((size_t)BN_ * HID_);
    float* ctx    = walloc((size_t)BN_ * HID_);
    float* outat  = walloc((size_t)BN_ * HID_);   // attention out (16,20480)
    float* sacc   = walloc((size_t)B_  * HID_);   // split-K accumulator

    dim3 blk(128, 1, 1);   // 4 waves/block, 1 wave per 16x16 tile

    // 1) fused mask statistics (streams the 64MB mask tensor once)
    obh_mask_stats<<<BN_, 256, 0, stream>>>(object_masks, target_mask, bboxes,
                                            edge, f, objsum);

    // 2) backbone hidden: h = relu(f*colsum(bb_w1) + bb_b1)
    obh_bb_hidden<<<(BN_*BBH_ + 255)/256, 256, 0, stream>>>(f, bb_w1, bb_b1, hbb);

    // 3) object_feats = h(320,2048) @ bb_fw(2048,1024) + bb_fb
    obh_gemm_wmma<0,false><<<dim3(HID_/64, BN_/16, 1), blk, 0, stream>>>(
        hbb, BBH_, bb_fw, HID_, bb_fb, nullptr, nullptr, nullptr, nullptr,
        feats, HID_, BN_, HID_, BBH_, BBH_);

    // 4) rel1 = relu(bn(edge(16,40) @ or_w1(40,1024) + or_b1))
    obh_gemm_wmma<1,false><<<dim3(HID_/64, B_/16, 1), blk, 0, stream>>>(
        edge, 2*N_, or_w1, HID_, or_b1, or_g, or_be, or_m, or_v,
        rel1, HID_, B_, HID_, 2*N_, 2*N_);

    // 5) rel = rel1(16,1024) @ or_w2(1024,20480) + or_b2  (84MB single-use: NT)
    obh_gemm_wmma<0,true><<<dim3((N_*HID_)/64, B_/16, 1), blk, 0, stream>>>(
        rel1, HID_, or_w2, N_*HID_, or_b2, nullptr, nullptr, nullptr, nullptr,
        rel, N_*HID_, B_, N_*HID_, HID_, HID_);

    // 6) q/k/v projections (320,1024)@(1024,1024); weights reused by 20 M-tiles -> RT
    obh_gemm_wmma<0,false><<<dim3(HID_/64, BN_/16, 1), blk, 0, stream>>>(
        feats, HID_, q_w, HID_, q_b, nullptr, nullptr, nullptr, nullptr,
        qb, HID_, BN_, HID_, HID_, HID_);
    obh_gemm_wmma<0,false><<<dim3(HID_/64, BN_/16, 1), blk, 0, stream>>>(
        rel, HID_, k_w, HID_, k_b, nullptr, nullptr, nullptr, nullptr,
        kb, HID_, BN_, HID_, HID_, HID_);
    obh_gemm_wmma<0,false><<<dim3(HID_/64, BN_/16, 1), blk, 0, stream>>>(
        rel, HID_, v_w, HID_, v_b, nullptr, nullptr, nullptr, nullptr,
        vb, HID_, BN_, HID_, HID_, HID_);

    // 7) attention (one block per batch*head)
    obh_attention<<<B_*HEADS_, 256, 0, stream>>>(qb, kb, vb, ctx);

    // 8) out = ctx(320,1024) @ o_w(1024,1024) + o_b
    obh_gemm_wmma<0,false><<<dim3(HID_/64, BN_/16, 1), blk, 0, stream>>>(
        ctx, HID_, o_w, HID_, o_b, nullptr, nullptr, nullptr, nullptr,
        outat, HID_, BN_, HID_, HID_, HID_);

    // 9) split-K: sacc += out(16,20480) @ at_w1(20480,1024)  (84MB single-use: NT)
    hipMemsetAsync(sacc, 0, (size_t)B_ * HID_ * sizeof(float), stream);
    obh_gemm_wmma<2,true><<<dim3(HID_/64, B_/16, 16), blk, 0, stream>>>(
        outat, N_*HID_, at_w1, HID_, nullptr, nullptr, nullptr, nullptr, nullptr,
        sacc, HID_, B_, HID_, N_*HID_, (N_*HID_)/16);

    // 10) s = relu(bn(sacc + at_b1))
    obh_bias_bn_relu<<<(B_*HID_ + 255)/256, 256, 0, stream>>>(
        sacc, at_b1, at_g, at_be, at_m, at_v, B_*HID_);

    // 11) scores = s @ at_w2 + at_b2, with padding select -> d_out (16,20)
    obh_final_scores<<<(BN_ + 7)/8, 256, 0, stream>>>(
        sacc, at_w2, at_b2, objsum, (float*)d_out);
}